// SPIGuidedGNN_24481313587799
// MI455X (gfx1250) — compile-verified
//
#include <hip/hip_runtime.h>
#include <hip/hip_bf16.h>

#define TAU_C 0.67f

typedef float v2f __attribute__((ext_vector_type(2)));
typedef float v8f __attribute__((ext_vector_type(8)));

// ---------------------------------------------------------------- utilities
__global__ void fill_f32(float* __restrict__ p, float v, size_t n) {
  size_t i = (size_t)blockIdx.x * blockDim.x + threadIdx.x;
  size_t st = (size_t)gridDim.x * blockDim.x;
  for (; i < n; i += st) p[i] = v;
}

__global__ void deg_accum(const int* __restrict__ dst, float* __restrict__ deg, int E) {
  int i = blockIdx.x * blockDim.x + threadIdx.x;
  int st = gridDim.x * blockDim.x;
  for (; i < E; i += st) atomicAdd(&deg[dst[i]], 1.0f);   // global_atomic_add_f32 (no return)
}

__global__ void rsqrt_k(const float* __restrict__ deg, float* __restrict__ dis, int n) {
  int i = blockIdx.x * blockDim.x + threadIdx.x;
  if (i < n) dis[i] = rsqrtf(deg[i]);
}

// ---------------------------------------------------------------- WMMA GEMM
// C[N,FOUT] = A[N,128] @ W[128,FOUT]  (+bias, optional relu, optional (1-beta) scale)
// fp32 path: V_WMMA_F32_16X16X4_F32, wave32.
// A fragment layout (ISA 7.12.2, 32-bit A 16x4): lane m=lane&15 holds row m;
//   lanes 0-15 -> K = {k0, k0+1}, lanes 16-31 -> K = {k0+2, k0+3}.
// B fragment (4x16): VGPR0 = (K=k0+koff, N=lane&15), VGPR1 = K=k0+koff+1, koff = lane<16?0:2.
// C/D: acc[r] -> row (r + (lane<16?0:8)), col = lane&15 within tile.
template<int FOUT, bool HAS_BIAS, bool RELU, bool GATE>
__launch_bounds__(256)
__global__ void gemm_wmma_f32(const float* __restrict__ A,
                              const float* __restrict__ W,
                              const float* __restrict__ bias,
                              const float* __restrict__ spi,
                              const float* __restrict__ logT,
                              float* __restrict__ C,
                              int N) {
  static_assert(FOUT % 16 == 0, "FOUT must be multiple of 16");
  __shared__ float wt[128 * FOUT];                 // <= 64 KB, identity layout
  for (int i = threadIdx.x; i < 128 * FOUT; i += 256) wt[i] = W[i];
  __syncthreads();

  const int lane    = threadIdx.x & 31;
  const int wave    = threadIdx.x >> 5;            // 0..7
  const int rowBase = blockIdx.x * 128 + wave * 16;
  const int m       = lane & 15;
  const int koff    = (lane < 16) ? 0 : 2;

  // Load full 16x128 A slab into registers (32 x v2f = 64 VGPRs), reused for all N-tiles.
  int rowL = rowBase + m; if (rowL > N - 1) rowL = N - 1;
  const float* arow = A + (size_t)rowL * 128 + koff;
  v2f af[32];
  #pragma unroll
  for (int ks = 0; ks < 32; ++ks) af[ks] = *(const v2f*)(arow + 4 * ks);

  float scale = 1.0f;
  if (GATE) {
    float T    = expf(logT[0]);
    float beta = 1.0f / (1.0f + expf(-(spi[0] - TAU_C) / T));
    scale = 1.0f - beta;
  }

  for (int nt = 0; nt < FOUT / 16; ++nt) {
    const int col = nt * 16 + m;
    v8f acc = {0.f, 0.f, 0.f, 0.f, 0.f, 0.f, 0.f, 0.f};
    #pragma unroll
    for (int ks = 0; ks < 32; ++ks) {
      v2f b;
      b.x = wt[(4 * ks + koff)     * FOUT + col];  // lane-stride-1 -> conflict-free
      b.y = wt[(4 * ks + koff + 1) * FOUT + col];
      acc = __builtin_amdgcn_wmma_f32_16x16x4_f32(false, af[ks], false, b,
                                                  (short)0, acc, false, false);
    }
    float bv = HAS_BIAS ? bias[col] : 0.0f;
    const int rOff = (lane < 16) ? 0 : 8;
    #pragma unroll
    for (int r = 0; r < 8; ++r) {
      int row = rowBase + r + rOff;
      if (row < N) {
        float v = acc[r] + bv;
        if (RELU) v = fmaxf(v, 0.0f);
        if (GATE) v *= scale;
        C[(size_t)row * FOUT + col] = v;
      }
    }
  }
}

// ---------------------------------------------------------------- edge scatter
// One wave per edge: agg[dst] += H[src] * dis[src]*dis[dst].  F/32 floats per lane.
template<int F>
__global__ void scatter_edges(const float* __restrict__ H, const int* __restrict__ src,
                              const int* __restrict__ dst, const float* __restrict__ dis,
                              float* __restrict__ agg, int E) {
  constexpr int VPL = F / 32;
  const int lane = threadIdx.x & 31;
  int w = (blockIdx.x * blockDim.x + threadIdx.x) >> 5;
  const int nw = (gridDim.x * blockDim.x) >> 5;
  for (int e = w; e < E; e += nw) {
    const int s = src[e];
    const int d = dst[e];
    const float nrm = dis[s] * dis[d];
    const float* hp = H   + (size_t)s * F + lane * VPL;
    float*       ap = agg + (size_t)d * F + lane * VPL;
    #pragma unroll
    for (int j = 0; j < VPL; ++j) atomicAdd(ap + j, hp[j] * nrm);
  }
}

// ---------------------------------------------------------------- epilogues
// z = relu(agg + H1/deg + bg1), in place on agg (self-loop norm = dis^2 = 1/deg)
__global__ void finalize_layer1(float* __restrict__ z, const float* __restrict__ H1,
                                const float* __restrict__ dis, const float* __restrict__ b,
                                int N) {
  size_t total = (size_t)N * 128;
  size_t i  = (size_t)blockIdx.x * blockDim.x + threadIdx.x;
  size_t st = (size_t)gridDim.x * blockDim.x;
  for (; i < total; i += st) {
    int node = (int)(i >> 7);
    int f    = (int)(i & 127);
    float ds = dis[node];
    float v  = z[i] + H1[i] * ds * ds + b[f];
    z[i] = fmaxf(v, 0.0f);
  }
}

// out += beta * (agg2 + H2/deg + bg2)   (out already holds (1-beta)*z_mlp)
__global__ void finalize_layer2(float* __restrict__ out, const float* __restrict__ agg2,
                                const float* __restrict__ H2, const float* __restrict__ dis,
                                const float* __restrict__ b, const float* __restrict__ spi,
                                const float* __restrict__ logT, int N) {
  float T    = expf(logT[0]);
  float beta = 1.0f / (1.0f + expf(-(spi[0] - TAU_C) / T));
  size_t total = (size_t)N * 64;
  size_t i  = (size_t)blockIdx.x * blockDim.x + threadIdx.x;
  size_t st = (size_t)gridDim.x * blockDim.x;
  for (; i < total; i += st) {
    int node = (int)(i >> 6);
    int f    = (int)(i & 63);
    float ds = dis[node];
    out[i] += beta * (agg2[i] + H2[i] * ds * ds + b[f]);
  }
}

// ---------------------------------------------------------------- launcher
extern "C" void kernel_launch(void* const* d_in, const int* in_sizes, int n_in,
                              void* d_out, int out_size, void* d_ws, size_t ws_size,
                              hipStream_t stream) {
  const float* x    = (const float*)d_in[0];
  const int*   ei   = (const int*)  d_in[1];
  const float* spi  = (const float*)d_in[2];
  const float* Wg1  = (const float*)d_in[3];
  const float* bg1  = (const float*)d_in[4];
  const float* Wg2  = (const float*)d_in[5];
  const float* bg2  = (const float*)d_in[6];
  const float* Wm1  = (const float*)d_in[7];
  const float* bm1  = (const float*)d_in[8];
  const float* Wm2  = (const float*)d_in[9];
  const float* bm2  = (const float*)d_in[10];
  const float* logT = (const float*)d_in[11];

  const int N = in_sizes[0] / 128;
  const int E = in_sizes[1] / 2;
  const int* src  = ei;
  const int* dstp = ei + E;
  float* out = (float*)d_out;

  // workspace: B0,B1,B2 = N*128 floats each; deg,dis = N floats (~155 MB total)
  float* B0  = (float*)d_ws;               // H1  -> later H2 (N*64)
  float* B1  = B0 + (size_t)N * 128;       // M1  -> later agg2 (N*64)
  float* B2  = B1 + (size_t)N * 128;       // agg -> z (in place)
  float* deg = B2 + (size_t)N * 128;
  float* dis = deg + N;

  const int gemmBlocks = (N + 127) / 128;

  // degrees (self-loop contributes the initial 1.0) and 1/sqrt(deg)
  fill_f32<<<1024, 256, 0, stream>>>(deg, 1.0f, (size_t)N);
  deg_accum<<<(E + 255) / 256, 256, 0, stream>>>(dstp, deg, E);
  rsqrt_k<<<(N + 255) / 256, 256, 0, stream>>>(deg, dis, N);

  // H1 = x @ Wg1 ; M1 = relu(x @ Wm1 + bm1)
  gemm_wmma_f32<128, false, false, false><<<gemmBlocks, 256, 0, stream>>>(x, Wg1, nullptr, nullptr, nullptr, B0, N);
  gemm_wmma_f32<128, true,  true,  false><<<gemmBlocks, 256, 0, stream>>>(x, Wm1, bm1,     nullptr, nullptr, B1, N);

  // layer-1 aggregation
  fill_f32<<<2048, 256, 0, stream>>>(B2, 0.0f, (size_t)N * 128);
  scatter_edges<128><<<4096, 256, 0, stream>>>(B0, src, dstp, dis, B2, E);
  finalize_layer1<<<2048, 256, 0, stream>>>(B2, B0, dis, bg1, N);

  // H2 = z @ Wg2 ; out = (1-beta) * (M1 @ Wm2 + bm2)
  gemm_wmma_f32<64, false, false, false><<<gemmBlocks, 256, 0, stream>>>(B2, Wg2, nullptr, nullptr, nullptr, B0, N);
  gemm_wmma_f32<64, true,  false, true ><<<gemmBlocks, 256, 0, stream>>>(B1, Wm2, bm2,     spi,     logT,    out, N);

  // layer-2 aggregation + fused gate
  fill_f32<<<2048, 256, 0, stream>>>(B1, 0.0f, (size_t)N * 64);
  scatter_edges<64><<<4096, 256, 0, stream>>>(B0, src, dstp, dis, B1, E);
  finalize_layer2<<<2048, 256, 0, stream>>>(out, B1, B0, dis, bg2, spi, logT, N);
}